// _fasterRCNN_73435350827078
// MI455X (gfx1250) — compile-verified
//
#include <hip/hip_runtime.h>
#include <hip/hip_bf16.h>
#include <math.h>

typedef __attribute__((ext_vector_type(2))) float v2f;
typedef __attribute__((ext_vector_type(8))) float v8f;

// ws layout (floats):
//   [0..21)   delta_margin accumulators
//   [32..53)  num_per_class accumulators
//   [64..85)  norm_weight result
//   [128 ..)  Wp2: W padded to 32 cols, pair-interleaved:
//             Wp2[(k>>1)*64 + n*2 + (k&1)] = W[k][n]  (n<21, else 0)

__global__ void _init_kernel(const float* __restrict__ W, float* __restrict__ Wp2,
                             float* __restrict__ stats, int D, int C) {
    int idx = blockIdx.x * blockDim.x + threadIdx.x;
    if (idx < 128) stats[idx] = 0.0f;
    int n  = idx & 31;
    int kp = idx >> 5;
    if (kp < (D >> 1)) {
        float lo = (n < C) ? W[(size_t)(2 * kp) * C + n] : 0.0f;
        float hi = (n < C) ? W[(size_t)(2 * kp + 1) * C + n] : 0.0f;
        Wp2[(size_t)kp * 64 + n * 2]     = lo;
        Wp2[(size_t)kp * 64 + n * 2 + 1] = hi;
    }
}

// One wave computes a 16-row x 32-col (21 valid) logit tile via V_WMMA_F32_16X16X4_F32.
// Block = 128 threads = 4 waves = 64 rows.
template <bool TGT>
__global__ __launch_bounds__(128)
void _gemm_softmax_kernel(const float* __restrict__ feat, const float* __restrict__ Wp2,
                          const float* __restrict__ bias, float* __restrict__ out,
                          float* __restrict__ g_dm, float* __restrict__ g_np,
                          int D, int C) {
    __shared__ float s_tile[64 * 32];   // 8 KB logits staging
    __shared__ float s_dm[32], s_np[32];

    const int tid  = threadIdx.x;
    const int lane = tid & 31;
    const int wave = tid >> 5;

    if (TGT && tid < 32) { s_dm[tid] = 0.0f; s_np[tid] = 0.0f; }

    const int mrow = lane & 15;          // output row within tile
    const int koff = (lane >> 4) << 1;   // 0 or 2: which K pair this lane holds
    const int r0   = blockIdx.x * 64 + wave * 16;

    const float* __restrict__ A = feat + (size_t)(r0 + mrow) * D + koff;
    // B fragment base: n*2 within a 64-float K-pair group; fold lane's K-pair in.
    const float* __restrict__ B = Wp2 + ((lane & 15) << 1) + ((koff >> 1) * 64);

    v8f acc0 = {0.f, 0.f, 0.f, 0.f, 0.f, 0.f, 0.f, 0.f};
    v8f acc1 = {0.f, 0.f, 0.f, 0.f, 0.f, 0.f, 0.f, 0.f};

    for (int k = 0; k < D; k += 16) {
#pragma unroll
        for (int j = 0; j < 4; ++j) {
            const int k4 = k + j * 4;
            v2f a = *(const v2f*)(A + k4);
            const float* bp = B + (size_t)(k4 >> 1) * 64;
            v2f b0 = *(const v2f*)(bp);        // N tile 0..15
            v2f b1 = *(const v2f*)(bp + 32);   // N tile 16..31 (zero-padded past C)
            acc0 = __builtin_amdgcn_wmma_f32_16x16x4_f32(false, a, false, b0,
                                                         (short)0, acc0, false, false);
            acc1 = __builtin_amdgcn_wmma_f32_16x16x4_f32(false, a, false, b1,
                                                         (short)0, acc1, false, false);
        }
    }

    // C/D layout: lane<16: VGPR i -> (M=i, N=lane); lane>=16: (M=8+i, N=lane-16)
    const int rb = wave * 16 + ((lane >> 4) << 3);
    const int cb = lane & 15;
#pragma unroll
    for (int i = 0; i < 8; ++i) {
        s_tile[(rb + i) * 32 + cb]      = acc0[i];
        s_tile[(rb + i) * 32 + cb + 16] = acc1[i];
    }
    __syncthreads();

    if (tid < 64) {
        const int grow = blockIdx.x * 64 + tid;
        float l[32];
        float lmax = -3.4e38f;
        int   larg = 0;
#pragma unroll
        for (int c = 0; c < 21; ++c) {
            float v = s_tile[tid * 32 + c] + bias[c];
            l[c] = v;
            if (v > lmax) { lmax = v; larg = c; }
        }
        float sumexp = 0.0f;
#pragma unroll
        for (int c = 0; c < 21; ++c) sumexp += expf(l[c] - lmax);
        float inv = 1.0f / sumexp;

        if (TGT) {
            // top1 = 1/sumexp; sum(probs)=1 -> mean_rest = (1-top1)/20
            float top1   = inv;
            float margin = top1 - (1.0f - top1) * (1.0f / 20.0f);
            if (larg != 0) {
                atomicAdd(&s_dm[larg], margin);
                atomicAdd(&s_np[larg], 1.0f);
            }
        } else {
            float* orow = out + (size_t)grow * 22;
#pragma unroll
            for (int c = 0; c < 21; ++c) orow[c] = expf(l[c] - lmax) * inv;
        }
    }

    if (TGT) {
        __syncthreads();
        if (tid < 21) {
            atomicAdd(&g_dm[tid], s_dm[tid]);
            atomicAdd(&g_np[tid], s_np[tid]);
        }
    }
}

__global__ void _finalize_kernel(const float* __restrict__ g_dm, const float* __restrict__ g_np,
                                 const float* __restrict__ tmf, const float* __restrict__ records,
                                 float* __restrict__ g_nw, int C) {
    if (threadIdx.x == 0) {
        float rec = records[0];
        float total = 0.0f;
        for (int c = 0; c < 21; ++c) total += g_np[c];
        float w[21];
        for (int c = 0; c < 21; ++c) {
            float blended = (tmf[c] * rec + g_dm[c] / (g_np[c] + 1e-6f)) / (rec + 1.0f);
            w[c] = (total > 0.0f) ? blended : tmf[c];
        }
        float mn = w[0], mx = w[0];
        for (int c = 1; c < 21; ++c) { mn = fminf(mn, w[c]); mx = fmaxf(mx, w[c]); }
        float d = 1.0f / (mx - mn + 1e-12f);
        for (int c = 0; c < 21; ++c) g_nw[c] = (w[c] - mn) * d;
    }
}

__global__ void _gather_kernel(const int* __restrict__ rois, const float* __restrict__ g_nw,
                               float* __restrict__ out, int S) {
    int i = blockIdx.x * blockDim.x + threadIdx.x;
    if (i < S) out[(size_t)i * 22 + 21] = g_nw[rois[i]];
}

extern "C" void kernel_launch(void* const* d_in, const int* in_sizes, int n_in,
                              void* d_out, int out_size, void* d_ws, size_t ws_size,
                              hipStream_t stream) {
    const float* src  = (const float*)d_in[0];
    const float* tgt  = (const float*)d_in[1];
    const float* Wc   = (const float*)d_in[2];
    const float* bc   = (const float*)d_in[3];
    const int*   rois = (const int*)d_in[4];
    const float* tmf  = (const float*)d_in[5];
    const float* rec  = (const float*)d_in[6];

    const int C = in_sizes[3];          // 21
    const int D = in_sizes[2] / C;      // 4096
    const int S = in_sizes[0] / D;      // 16384
    const int T = in_sizes[1] / D;      // 32768

    float* ws   = (float*)d_ws;
    float* g_dm = ws;
    float* g_np = ws + 32;
    float* g_nw = ws + 64;
    float* Wp2  = ws + 128;             // (D/2)*64 floats = 512 KB

    float* out = (float*)d_out;

    const int initThreads = (D / 2) * 32;
    _init_kernel<<<(initThreads + 255) / 256, 256, 0, stream>>>(Wc, Wp2, ws, D, C);

    _gemm_softmax_kernel<true ><<<T / 64, 128, 0, stream>>>(tgt, Wp2, bc, nullptr, g_dm, g_np, D, C);
    _gemm_softmax_kernel<false><<<S / 64, 128, 0, stream>>>(src, Wp2, bc, out,     g_dm, g_np, D, C);

    _finalize_kernel<<<1, 32, 0, stream>>>(g_dm, g_np, tmf, rec, g_nw, C);
    _gather_kernel<<<(S + 255) / 256, 256, 0, stream>>>(rois, g_nw, out, S);
}